// LightGCNEncoder_46651934769921
// MI455X (gfx1250) — compile-verified
//
#include <hip/hip_runtime.h>

typedef float v2f __attribute__((ext_vector_type(2)));
typedef float v8f __attribute__((ext_vector_type(8)));

// Native non-returning FP32 global atomic add (GLOBAL_ATOMIC_ADD_F32).
// atomicAdd(float*) would lower to a CAS retry loop without
// -munsafe-fp-atomics; unsafeAtomicAdd maps to the hardware instruction.
__device__ __forceinline__ void gatomic_add(float* p, float v) {
  unsafeAtomicAdd(p, v);
}

// ---------------------------------------------------------------------------
// Kernel 1: deg[i] = 1.0  (self-loop contribution to degree)
// ---------------------------------------------------------------------------
__global__ void k_deg_init(float* __restrict__ deg, int n) {
  int i = blockIdx.x * blockDim.x + threadIdx.x;
  if (i < n) deg[i] = 1.0f;
}

// ---------------------------------------------------------------------------
// Kernel 2: deg[dst[e]] += 1 over real edges
// ---------------------------------------------------------------------------
__global__ void k_deg_accum(const int* __restrict__ dst, float* __restrict__ deg, int e) {
  int i = blockIdx.x * blockDim.x + threadIdx.x;
  if (i < e) gatomic_add(&deg[dst[i]], 1.0f);
}

// ---------------------------------------------------------------------------
// Kernel 3: dinv[i] = deg>0 ? rsqrt(deg) : 0   (in place)
// ---------------------------------------------------------------------------
__global__ void k_dinv(float* __restrict__ deg, int n) {
  int i = blockIdx.x * blockDim.x + threadIdx.x;
  if (i < n) {
    float d = deg[i];
    deg[i] = (d > 0.0f) ? rsqrtf(d) : 0.0f;
  }
}

// ---------------------------------------------------------------------------
// Kernel 4: fused dual GEMM via V_WMMA_F32_16X16X4_F32.
//   xw_mu = x @ Wmu ; xw_lv = x @ Wlv   (stored to workspace)
//   out_* = xw_* * dinv[m]^2 + b_*      (self-loop term seeds the output)
// One wave32 per 16-row strip; 4 column tiles x 2 weight matrices.
// N % 16 == 0, so each wave is entirely valid -> EXEC all-1s across every
// WMMA as the ISA requires.
// ---------------------------------------------------------------------------
__global__ __launch_bounds__(128)
void k_gemm_wmma(const float* __restrict__ x,
                 const float* __restrict__ Wmu, const float* __restrict__ bmu,
                 const float* __restrict__ Wlv, const float* __restrict__ blv,
                 const float* __restrict__ dinv,
                 float* __restrict__ xw_mu, float* __restrict__ xw_lv,
                 float* __restrict__ out_mu, float* __restrict__ out_lv,
                 int n) {
  const int lane = threadIdx.x & 31;
  const int wave = threadIdx.x >> 5;
  const int m0 = (blockIdx.x * 4 + wave) * 16;
  if (m0 >= n) return;                 // whole-wave exit only (16 | n)

  const int h  = lane >> 4;            // half-wave: selects K pair / M+8
  const int ln = lane & 15;            // 0..15: A row / B,C,D column

  v8f acc_mu[4] = {{0,0,0,0,0,0,0,0}, {0,0,0,0,0,0,0,0},
                   {0,0,0,0,0,0,0,0}, {0,0,0,0,0,0,0,0}};
  v8f acc_lv[4] = {{0,0,0,0,0,0,0,0}, {0,0,0,0,0,0,0,0},
                   {0,0,0,0,0,0,0,0}, {0,0,0,0,0,0,0,0}};

  const float* xrow = x + (size_t)(m0 + ln) * 64;

  #pragma unroll
  for (int kt = 0; kt < 16; ++kt) {    // K = 64, 4 per WMMA
    const int kbase = kt * 4 + h * 2;  // even -> 8B-aligned float2 load
    const v2f a = *(const v2f*)(xrow + kbase);
    #pragma unroll
    for (int t = 0; t < 4; ++t) {      // 4 column tiles of 16
      const int col = t * 16 + ln;
      v2f bm, bl;
      bm.x = Wmu[(size_t)(kbase + 0) * 64 + col];
      bm.y = Wmu[(size_t)(kbase + 1) * 64 + col];
      bl.x = Wlv[(size_t)(kbase + 0) * 64 + col];
      bl.y = Wlv[(size_t)(kbase + 1) * 64 + col];
      acc_mu[t] = __builtin_amdgcn_wmma_f32_16x16x4_f32(
          false, a, false, bm, (short)0, acc_mu[t], false, false);
      acc_lv[t] = __builtin_amdgcn_wmma_f32_16x16x4_f32(
          false, a, false, bl, (short)0, acc_lv[t], false, false);
    }
  }

  // D layout: VGPR r <-> row (r + 8h), column ln within each 16-col tile.
  #pragma unroll
  for (int r = 0; r < 8; ++r) {
    const int m = m0 + r + 8 * h;
    const float di = dinv[m];
    const float selfnorm = di * di;
    #pragma unroll
    for (int t = 0; t < 4; ++t) {
      const int c = t * 16 + ln;
      const size_t idx = (size_t)m * 64 + c;
      const float vmu = acc_mu[t][r];
      const float vlv = acc_lv[t][r];
      xw_mu[idx]  = vmu;
      xw_lv[idx]  = vlv;
      out_mu[idx] = vmu * selfnorm + bmu[c];
      out_lv[idx] = vlv * selfnorm + blv[c];
    }
  }
}

// ---------------------------------------------------------------------------
// Kernel 5: wave-per-edge scatter-add.
//   norm = dinv[src]*dinv[dst];  out[dst] += xw[src] * norm  (both matrices)
// Lane l handles channels 2l, 2l+1 via one float2 gather per matrix
// (256B/wave, fully coalesced); adds go out as native non-returning
// global_atomic_add_f32 (STOREcnt-tracked, no retry loop).
// ---------------------------------------------------------------------------
__global__ __launch_bounds__(256)
void k_scatter(const int* __restrict__ src, const int* __restrict__ dst,
               const float* __restrict__ dinv,
               const float* __restrict__ xw_mu, const float* __restrict__ xw_lv,
               float* __restrict__ out_mu, float* __restrict__ out_lv,
               int e) {
  const int lane = threadIdx.x & 31;
  const int edge = blockIdx.x * (blockDim.x >> 5) + (threadIdx.x >> 5);
  if (edge >= e) return;
  const int s = src[edge];
  const int d = dst[edge];
  const float w = dinv[s] * dinv[d];
  const size_t sb = (size_t)s * 64 + 2 * lane;
  const size_t db = (size_t)d * 64 + 2 * lane;

  const v2f xm = *(const v2f*)(xw_mu + sb);   // 8B-aligned (rows 256B-aligned)
  const v2f xl = *(const v2f*)(xw_lv + sb);
  gatomic_add(&out_mu[db + 0], xm.x * w);
  gatomic_add(&out_mu[db + 1], xm.y * w);
  gatomic_add(&out_lv[db + 0], xl.x * w);
  gatomic_add(&out_lv[db + 1], xl.y * w);
}

// ---------------------------------------------------------------------------
extern "C" void kernel_launch(void* const* d_in, const int* in_sizes, int n_in,
                              void* d_out, int out_size, void* d_ws, size_t ws_size,
                              hipStream_t stream) {
  const float* x    = (const float*)d_in[0];
  const int*   ei   = (const int*)d_in[1];
  const float* Wmu  = (const float*)d_in[2];
  const float* bmu  = (const float*)d_in[3];
  const float* Wlv  = (const float*)d_in[4];
  const float* blv  = (const float*)d_in[5];

  const int n = in_sizes[0] / 64;      // nodes
  const int e = in_sizes[1] / 2;       // edges
  const int* src = ei;                 // edge_index[0]
  const int* dst = ei + e;             // edge_index[1]

  // Workspace layout (all fp32), 256B-aligned sections:
  //   [0)      dinv  : n floats
  //   [off1)   xw_mu : n*64 floats
  //   [off2)   xw_lv : n*64 floats
  char* ws = (char*)d_ws;
  size_t off1 = ((size_t)n * sizeof(float) + 255) & ~(size_t)255;
  size_t off2 = off1 + (((size_t)n * 64 * sizeof(float) + 255) & ~(size_t)255);
  float* dinv  = (float*)ws;
  float* xw_mu = (float*)(ws + off1);
  float* xw_lv = (float*)(ws + off2);

  float* out_mu = (float*)d_out;
  float* out_lv = out_mu + (size_t)n * 64;

  const int T = 256;
  // 1) degree (self-loops) -> 2) accumulate edges -> 3) rsqrt in place
  k_deg_init <<<(n + T - 1) / T, T, 0, stream>>>(dinv, n);
  k_deg_accum<<<(e + T - 1) / T, T, 0, stream>>>(dst, dinv, e);
  k_dinv     <<<(n + T - 1) / T, T, 0, stream>>>(dinv, n);

  // 4) fused dual WMMA GEMM + self-loop/bias seeding of outputs
  k_gemm_wmma<<<(n + 63) / 64, 128, 0, stream>>>(
      x, Wmu, bmu, Wlv, blv, dinv, xw_mu, xw_lv, out_mu, out_lv, n);

  // 5) edge scatter-add (8 waves / block, one edge per wave)
  k_scatter<<<(e + 7) / 8, 256, 0, stream>>>(
      src, dst, dinv, xw_mu, xw_lv, out_mu, out_lv, e);
}